// CustomLoss_38276748542062
// MI455X (gfx1250) — compile-verified
//
#include <hip/hip_runtime.h>
#include <cstdint>

#define Bdim 32
#define Tdim 256
#define Vdim 32000

// 31 full async chunks of 1024 floats (256 thr * float4) + 256-float remainder
#define NCHUNK 31
#define CHUNK_FLOATS 1024
#define REM_BASE (NCHUNK * CHUNK_FLOATS) // 31744
#define REM_F4 64                        // (32000-31744)/4
#define NBUF 4                           // async pipeline depth (16KB LDS)

__device__ __forceinline__ float exp2_fast(float x) { return __builtin_amdgcn_exp2f(x); }
__device__ __forceinline__ float log2_fast(float x) { return __builtin_amdgcn_logf(x); }

__device__ __forceinline__ void online_update(const float4 v, float& m, float& s) {
  const float L2E = 1.4426950408889634f;
  float c  = fmaxf(fmaxf(v.x, v.y), fmaxf(v.z, v.w));
  float mn = fmaxf(m, c);
  s = s * exp2_fast((m - mn) * L2E)
    + exp2_fast((v.x - mn) * L2E)
    + exp2_fast((v.y - mn) * L2E)
    + exp2_fast((v.z - mn) * L2E)
    + exp2_fast((v.w - mn) * L2E);
  m = mn;
}

__device__ __forceinline__ void combine(float& m, float& s, float m2, float s2) {
  const float L2E = 1.4426950408889634f;
  float mn = fmaxf(m, m2);
  s = s * exp2_fast((m - mn) * L2E) + s2 * exp2_fast((m2 - mn) * L2E);
  m = mn;
}

__device__ __forceinline__ void async_load_f4(uint32_t lds_off, uint64_t gaddr) {
  // Non-temporal: 1GB stream read exactly once, keep it out of L2.
  asm volatile("global_load_async_to_lds_b128 %0, %1, off th:TH_LOAD_NT"
               :: "v"(lds_off), "v"(gaddr) : "memory");
}

// One workgroup (256 threads = 8 wave32) per (b,t) row of 32000 logits.
// Streams the row through LDS with a 4-deep async global->LDS pipeline.
__global__ void __launch_bounds__(256)
ce_row_lse_kernel(const float* __restrict__ pred,
                  const int* __restrict__ gt,
                  const int* __restrict__ len,
                  float* __restrict__ row_nll) {
  const int row = blockIdx.x;
  const int b = row >> 8;        // T = 256
  const int t = row & 255;
  const int tid = threadIdx.x;

  // Mask check first: skip the whole 128KB row if this position is masked
  // (~50% of rows for uniform lengths -> halves HBM traffic).
  const int tgt_len = len[b] - 1;
  if (t >= tgt_len) {
    if (tid == 0) row_nll[row] = 0.0f;
    return;
  }

  const float* rowp = pred + (size_t)row * Vdim;

  __shared__ float4 stage[NBUF][256];
  __shared__ float red_m[8];
  __shared__ float red_s[8];

  // Target logit gather (issued early; latency hidden behind the stream).
  float x_t = 0.0f;
  if (tid == 0) x_t = rowp[gt[row]];

  // Flat shared-aperture address keeps the LDS byte offset in its low 32 bits.
  const uint32_t lds_base = (uint32_t)(uintptr_t)&stage[0][tid];
  const float4* gp = (const float4*)rowp + tid; // chunk stride = 256 float4

  // Prime the pipe: chunks 0..NBUF-2 into buffers 0..NBUF-2.
  #pragma unroll
  for (int p = 0; p < NBUF - 1; ++p) {
    async_load_f4(lds_base + (uint32_t)p * 4096u,
                  (uint64_t)(uintptr_t)(gp + (size_t)p * 256));
  }

  float m = -3.402823466e38f;
  float s = 0.0f;

  for (int c = 0; c < NCHUNK; ++c) {
    if (c + NBUF - 1 < NCHUNK) {
      async_load_f4(lds_base + (uint32_t)((c + NBUF - 1) & (NBUF - 1)) * 4096u,
                    (uint64_t)(uintptr_t)(gp + (size_t)(c + NBUF - 1) * 256));
      // NBUF in flight; ensure the oldest (chunk c) has landed in LDS.
      asm volatile("s_wait_asynccnt 0x3" ::: "memory");
    } else {
      // Tail: remaining chunks are all in flight already.
      asm volatile("s_wait_asynccnt 0x0" ::: "memory");
    }
    float4 v = stage[c & (NBUF - 1)][tid]; // read back own staged data
    online_update(v, m, s);
  }

  // Remainder: 256 floats = 64 float4, direct global loads.
  if (tid < REM_F4) {
    float4 v = ((const float4*)(rowp + REM_BASE))[tid];
    online_update(v, m, s);
  }

  // Wave32 reduction of (m, s).
  #pragma unroll
  for (int off = 16; off >= 1; off >>= 1) {
    float m2 = __shfl_xor(m, off, 32);
    float s2 = __shfl_xor(s, off, 32);
    combine(m, s, m2, s2);
  }
  const int wave = tid >> 5;
  if ((tid & 31) == 0) { red_m[wave] = m; red_s[wave] = s; }
  __syncthreads();

  if (tid == 0) {
    float M = red_m[0], S = red_s[0];
    #pragma unroll
    for (int w = 1; w < 8; ++w) combine(M, S, red_m[w], red_s[w]);
    const float LN2 = 0.6931471805599453f;
    float lse = M + log2_fast(S) * LN2;
    row_nll[row] = lse - x_t;
  }
}

// Deterministic final reduction: sum 8192 per-row nll values + valid count.
__global__ void __launch_bounds__(256)
ce_finalize_kernel(const float* __restrict__ row_nll,
                   const int* __restrict__ len,
                   float* __restrict__ out) {
  __shared__ float red[8];
  const int tid = threadIdx.x;

  float s = 0.0f;
  #pragma unroll
  for (int i = 0; i < (Bdim * Tdim) / 256; ++i)
    s += row_nll[tid + i * 256];

  #pragma unroll
  for (int off = 16; off >= 1; off >>= 1)
    s += __shfl_xor(s, off, 32);
  if ((tid & 31) == 0) red[tid >> 5] = s;
  __syncthreads();

  if (tid == 0) {
    float total = 0.0f;
    #pragma unroll
    for (int w = 0; w < 8; ++w) total += red[w];
    int cnt = 0;
    for (int bb = 0; bb < Bdim; ++bb) {
      int tl = len[bb] - 1;
      if (tl > 0) cnt += (tl < Tdim ? tl : Tdim);
    }
    out[0] = total / (float)(cnt > 0 ? cnt : 1);
  }
}

extern "C" void kernel_launch(void* const* d_in, const int* in_sizes, int n_in,
                              void* d_out, int out_size, void* d_ws, size_t ws_size,
                              hipStream_t stream) {
  const float* pred = (const float*)d_in[0];   // [32,256,32000] f32
  const int*   gt   = (const int*)d_in[1];     // [32,256]
  const int*   len  = (const int*)d_in[2];     // [32]
  float* out = (float*)d_out;                  // scalar f32
  float* row_nll = (float*)d_ws;               // 8192 floats of scratch

  ce_row_lse_kernel<<<Bdim * Tdim, 256, 0, stream>>>(pred, gt, len, row_nll);
  ce_finalize_kernel<<<1, 256, 0, stream>>>(row_nll, len, out);
}